// HuffmanTree_3917010174472
// MI455X (gfx1250) — compile-verified
//
#include <hip/hip_runtime.h>
#include <math.h>
#include <stdint.h>

// CDNA5 / gfx1250: wave32, WMMA (not MFMA), async global->LDS DMA path.
typedef __attribute__((ext_vector_type(2))) float v2f;  // A/B of V_WMMA_F32_16X16X4_F32 (16x4 f32 = 2 VGPR)
typedef __attribute__((ext_vector_type(8))) float v8f;  // 16x16 f32 C/D = 8 VGPR

#define LPATH 24
#define DIM 1024
#define ROWSTRIDE 1028  // +4 floats pad: WMMA A-operand ds_load_b64 hits all 64 banks; 4112B = 16B-aligned

__global__ __launch_bounds__(256) void huffman_wmma_kernel(
    const float* __restrict__ word_vec,      // [B, DIM]
    const float* __restrict__ node_vectors,  // [N, DIM]
    const int*   __restrict__ node_ids,      // [B, LPATH]
    const int*   __restrict__ digits,        // [B, LPATH]
    const int*   __restrict__ path_len,      // [B]
    float* __restrict__ out,                 // [B]
    int B)
{
    __shared__ float rows_s[LPATH * ROWSTRIDE];  // 98688 B: 24 gathered node rows
    __shared__ float wv_s[DIM];                  // 4096 B: word_vec[b]
    __shared__ float red[8][16];                 // per-wave partial scores
    __shared__ float pvals[LPATH];
    __shared__ int   ids_s[LPATH];
    __shared__ int   dig_s[LPATH];

    const int b = blockIdx.x;
    if (b >= B) return;
    const int tid  = threadIdx.x;
    const int wave = tid >> 5;
    const int lane = tid & 31;

    if (tid < LPATH) {
        ids_s[tid] = node_ids[b * LPATH + tid];
        dig_s[tid] = digits[b * LPATH + tid];
    }

    // word_vec[b] -> LDS via async DMA: 256 threads x b128 = 4096 B, no VGPR round-trip.
    {
        uint32_t ldsoff = (uint32_t)(uintptr_t)wv_s + (uint32_t)tid * 16u;
        unsigned long long gaddr =
            (unsigned long long)(const char*)(word_vec + (size_t)b * DIM) + (unsigned long long)tid * 16ull;
        asm volatile("global_load_async_to_lds_b128 %0, %1, off offset:0"
                     :: "v"(ldsoff), "v"(gaddr) : "memory");
    }
    __syncthreads();  // ids_s visible (async wv still in flight; waited below)

    // Gather 24 node rows (4 KB each) into LDS with async DMA.
    // Wave w owns rows {w, w+8, w+16}; per row: 32 lanes x 16B x 8 insts = 4 KB.
    for (int j = 0; j < 3; ++j) {
        const int r = wave + j * 8;
        uint32_t ldsoff = (uint32_t)(uintptr_t)(rows_s + r * ROWSTRIDE) + (uint32_t)lane * 16u;
        unsigned long long gaddr =
            (unsigned long long)(const char*)(node_vectors + (size_t)ids_s[r] * DIM) +
            (unsigned long long)lane * 16ull;
        asm volatile(
            "global_load_async_to_lds_b128 %0, %1, off offset:0\n\t"
            "global_load_async_to_lds_b128 %0, %1, off offset:512\n\t"
            "global_load_async_to_lds_b128 %0, %1, off offset:1024\n\t"
            "global_load_async_to_lds_b128 %0, %1, off offset:1536\n\t"
            "global_load_async_to_lds_b128 %0, %1, off offset:2048\n\t"
            "global_load_async_to_lds_b128 %0, %1, off offset:2560\n\t"
            "global_load_async_to_lds_b128 %0, %1, off offset:3072\n\t"
            "global_load_async_to_lds_b128 %0, %1, off offset:3584"
            :: "v"(ldsoff), "v"(gaddr) : "memory");
    }
    asm volatile("s_wait_asynccnt 0x0" ::: "memory");  // my async copies landed in LDS
    __syncthreads();                                    // everyone's copies visible

    // ---- WMMA phase: scores[l] = dot(rows[l], wv), via V_WMMA_F32_16X16X4_F32 ----
    // M = l within tile (2 tiles of 16 over L=24), K = 4 d-values, N = replicated wv.
    const int tile  = wave & 1;   // l-tile: 0 -> l=0..15, 1 -> l=16..23 (+pad)
    const int chunk = wave >> 1;  // d-chunk: [chunk*256, chunk*256+256)
    const int m     = lane & 15;
    const int grp   = lane >> 4;  // lane group: K={0,1} vs K={2,3}

    int r = tile * 16 + m;
    if (r >= LPATH) r = 0;        // padded rows: computed then discarded

    const float* arow = rows_s + r * ROWSTRIDE + chunk * 256 + grp * 2;
    const float* brow = wv_s + chunk * 256 + grp * 2;

    v8f c0 = {0.f, 0.f, 0.f, 0.f, 0.f, 0.f, 0.f, 0.f};
    v8f c1 = c0;
#pragma unroll 4
    for (int it = 0; it < 32; ++it) {  // 2 accumulators x 32 iters x K=4 = 256 d per wave
        v2f a0 = *(const v2f*)(arow + it * 8);
        v2f b0 = *(const v2f*)(brow + it * 8);
        v2f a1 = *(const v2f*)(arow + it * 8 + 4);
        v2f b1 = *(const v2f*)(brow + it * 8 + 4);
        c0 = __builtin_amdgcn_wmma_f32_16x16x4_f32(false, a0, false, b0,
                                                   (short)0, c0, false, false);
        c1 = __builtin_amdgcn_wmma_f32_16x16x4_f32(false, a1, false, b1,
                                                   (short)0, c1, false, false);
    }
    v8f c = c0 + c1;

    // All N columns equal (B replicated): lane 0 holds M=0..7, lane 16 holds M=8..15.
    if (lane == 0 || lane == 16) {
        const int mb = grp * 8;
#pragma unroll
        for (int i = 0; i < 8; ++i) red[wave][mb + i] = c[i];
    }
    __syncthreads();

    // ---- Combine d-chunks, sigmoid with digit sign, mask, product ----
    if (tid < LPATH) {
        const int l  = tid;
        const int t  = l >> 4;
        const int mm = l & 15;
        float s = red[t][mm] + red[t + 2][mm] + red[t + 4][mm] + red[t + 6][mm];
        float x = (dig_s[l] == 1) ? s : -s;
        float p = 1.0f / (1.0f + __expf(-x));
        if (l >= path_len[b]) p = 1.0f;
        pvals[l] = p;
    }
    __syncthreads();
    if (tid == 0) {
        float prod = 1.0f;
#pragma unroll
        for (int i = 0; i < LPATH; ++i) prod *= pvals[i];
        out[b] = prod;
    }
}

extern "C" void kernel_launch(void* const* d_in, const int* in_sizes, int n_in,
                              void* d_out, int out_size, void* d_ws, size_t ws_size,
                              hipStream_t stream) {
    const float* word_vec     = (const float*)d_in[0];
    const float* node_vectors = (const float*)d_in[1];
    const int*   node_ids     = (const int*)d_in[2];
    const int*   digits       = (const int*)d_in[3];
    const int*   path_len     = (const int*)d_in[4];
    float* out = (float*)d_out;
    const int B = in_sizes[4];  // path_len has one entry per batch element

    hipLaunchKernelGGL(huffman_wmma_kernel, dim3(B), dim3(256), 0, stream,
                       word_vec, node_vectors, node_ids, digits, path_len, out, B);
}